// EncoderNTM_90709709291816
// MI455X (gfx1250) — compile-verified
//
#include <hip/hip_runtime.h>
#include <math.h>
#include <stdint.h>

#define IDIM   256
#define CDIM   512
#define NHEADS 2
#define NSLOT  128   // N
#define MW     64    // M
#define T_STEPS 64
#define BSZ    128

#define BPW     16            // batch rows per workgroup (= WMMA M tile)
#define NWG     (BSZ/BPW)     // 8 workgroups
#define NTHREADS 512
#define NWAVES  16

#define XDIM  (IDIM + NHEADS*MW)   // 384
#define ZDIM  (XDIM + CDIM)        // 896
#define GDIM  (4*CDIM)             // 2048
#define RCOLS 80                   // M+6=70 padded to 5 tiles
#define WCOLS 208                  // 3M+6=198 padded to 13 tiles
#define ODIM  (CDIM + NHEADS*MW)   // 640

// padded LDS row strides. ZS=900: rows are 16B-aligned (3600 B) for async
// B128 LDS writes, and 900 mod 64 banks = 4 -> the 16 A-rows map to 16
// distinct banks for the WMMA A-operand loads.
#define ZS   900
#define GS   2049
#define CS   513
#define HOS  209
#define SIMS 129

typedef __attribute__((ext_vector_type(2))) float v2f;
typedef __attribute__((ext_vector_type(8))) float v8f;

__device__ __forceinline__ float sigmoidf_(float x) { return 1.0f / (1.0f + __expf(-x)); }
__device__ __forceinline__ float softplusf_(float x) { return (x > 20.0f) ? x : log1pf(__expf(x)); }

__device__ __forceinline__ float wave_sum(float v) {
  for (int m = 16; m >= 1; m >>= 1) v += __shfl_xor(v, m, 32);
  return v;
}
__device__ __forceinline__ float wave_max(float v) {
  for (int m = 16; m >= 1; m >>= 1) v = fmaxf(v, __shfl_xor(v, m, 32));
  return v;
}

// Async DMA: 16 bytes per lane, global -> LDS, tracked by ASYNCcnt.
// Low 32 bits of a generic LDS-aperture address are the LDS byte offset.
__device__ __forceinline__ void async_g2l_b128(const float* gaddr, float* laddr) {
  unsigned lofs = (unsigned)(uintptr_t)laddr;
  unsigned long long ga = (unsigned long long)(uintptr_t)gaddr;
  asm volatile("global_load_async_to_lds_b128 %0, %1, off"
               :: "v"(lofs), "v"(ga)
               : "memory");
}

__device__ __forceinline__ void wait_async0() {
#if __has_builtin(__builtin_amdgcn_s_wait_asynccnt)
  __builtin_amdgcn_s_wait_asynccnt(0);
#else
  asm volatile("s_wait_asynccnt 0x0" ::: "memory");
#endif
}

// One 16x16 output tile, f32 WMMA, K in steps of 4 with two independent
// accumulator chains (klen must be a multiple of 8).
// A: [16 rows][lda] (LDS), B: [K][ldb] (global, pre-transposed K-major).
__device__ __forceinline__ v8f wmma_gemm(const float* A, int lda,
                                         const float* Bt, int ldb,
                                         int klen, int colbase) {
  const int lane = threadIdx.x & 31;
  const int half = lane >> 4;
  const int m    = lane & 15;
  v8f acc0 = {}, acc1 = {};
  const float* arow = A + m * lda;
  const float* bcol = Bt + colbase + m;
  for (int kb = 0; kb < klen; kb += 8) {
    const int k0 = kb + 2 * half;
    v2f a0, b0, a1, b1;
    a0.x = arow[k0];
    a0.y = arow[k0 + 1];
    b0.x = bcol[(size_t)k0 * ldb];
    b0.y = bcol[(size_t)(k0 + 1) * ldb];
    acc0 = __builtin_amdgcn_wmma_f32_16x16x4_f32(false, a0, false, b0,
                                                 (short)0, acc0, false, false);
    const int k1 = k0 + 4;
    a1.x = arow[k1];
    a1.y = arow[k1 + 1];
    b1.x = bcol[(size_t)k1 * ldb];
    b1.y = bcol[(size_t)(k1 + 1) * ldb];
    acc1 = __builtin_amdgcn_wmma_f32_16x16x4_f32(false, a1, false, b1,
                                                 (short)0, acc1, false, false);
  }
  return acc0 + acc1;
}

// NTM addressing (one wave handles one batch element). hob = head output row
// (k | beta | g | s0 s1 s2 | gamma ...). wprev[NSLOT] updated in place to the
// new weighting. simrow/wgrow are per-batch LDS scratch rows.
__device__ __forceinline__ void address_head(float* memb, const float* hob,
                                             float* wprev, float* simrow,
                                             float* wgrow) {
  const int lane = threadIdx.x & 31;
  const float beta  = softplusf_(hob[MW + 0]);
  const float g     = sigmoidf_(hob[MW + 1]);
  const float s0r = hob[MW + 2], s1r = hob[MW + 3], s2r = hob[MW + 4];
  const float smax = fmaxf(s0r, fmaxf(s1r, s2r));
  const float e0 = __expf(s0r - smax), e1 = __expf(s1r - smax), e2 = __expf(s2r - smax);
  const float einv = 1.0f / (e0 + e1 + e2);
  const float s0 = e0 * einv, s1 = e1 * einv, s2 = e2 * einv;
  const float gamma = 1.0f + softplusf_(hob[MW + 5]);

  // ||k + eps||
  float kv0 = hob[lane] + 1e-16f;
  float kv1 = hob[lane + 32] + 1e-16f;
  float kn  = sqrtf(wave_sum(kv0 * kv0 + kv1 * kv1));

  // beta * cosine similarity per slot
  for (int n = lane; n < NSLOT; n += 32) {
    const float* mrow = memb + n * MW;
    float dot = 0.0f, mn = 0.0f;
    for (int m2 = 0; m2 < MW; ++m2) {
      float mv = mrow[m2] + 1e-16f;
      float kv = hob[m2] + 1e-16f;
      dot += mv * kv;
      mn  += mv * mv;
    }
    float denom = fmaxf(sqrtf(mn) * kn, 1e-8f);
    simrow[n] = beta * (dot / denom);
  }
  // softmax over slots (wave-local)
  float mx = -1e30f;
  for (int n = lane; n < NSLOT; n += 32) mx = fmaxf(mx, simrow[n]);
  mx = wave_max(mx);
  float sum = 0.0f;
  for (int n = lane; n < NSLOT; n += 32) {
    float e = __expf(simrow[n] - mx);
    simrow[n] = e;
    sum += e;
  }
  sum = wave_sum(sum);
  const float sinv = 1.0f / sum;
  // interpolate with previous weighting
  for (int n = lane; n < NSLOT; n += 32) {
    float wc = simrow[n] * sinv;
    wgrow[n] = g * wc + (1.0f - g) * wprev[n];
  }
  // circular shift-convolution + sharpening (same-wave DS ops stay in order)
  float psum = 0.0f;
  for (int n = lane; n < NSLOT; n += 32) {
    float wm1 = wgrow[(n + NSLOT - 1) & (NSLOT - 1)];
    float w0  = wgrow[n];
    float wp1 = wgrow[(n + 1) & (NSLOT - 1)];
    float wv = s0 * wm1 + s1 * w0 + s2 * wp1;
    float wp = __powf(wv, gamma);
    simrow[n] = wp;
    psum += wp;
  }
  psum = wave_sum(psum);
  const float pinv = 1.0f / (psum + 1e-16f);
  for (int n = lane; n < NSLOT; n += 32) wprev[n] = simrow[n] * pinv;
}

// Pre-pass: build K-major weight copies so GEMM B-loads coalesce across
// lanes: WcatT[k][n] = [W_ih | W_hh].T, WrT/WwT zero-padded to tile width.
__global__ void ntm_prep_kernel(const float* __restrict__ W_ih,
                                const float* __restrict__ W_hh,
                                const float* __restrict__ Wr,
                                const float* __restrict__ Ww,
                                float* __restrict__ WcatT,
                                float* __restrict__ WrT,
                                float* __restrict__ WwT) {
  size_t i = (size_t)blockIdx.x * blockDim.x + threadIdx.x;
  const size_t NCAT = (size_t)ZDIM * GDIM;
  const size_t NR   = (size_t)NHEADS * CDIM * RCOLS;
  const size_t NWW  = (size_t)NHEADS * CDIM * WCOLS;
  if (i < NCAT) {
    int k = (int)(i / GDIM), n = (int)(i % GDIM);
    WcatT[i] = (k < XDIM) ? W_ih[(size_t)n * XDIM + k]
                          : W_hh[(size_t)n * CDIM + (k - XDIM)];
  } else if (i < NCAT + NR) {
    size_t j = i - NCAT;
    int hi = (int)(j / (CDIM * RCOLS));
    int r  = (int)(j % (CDIM * RCOLS));
    int k = r / RCOLS, c = r % RCOLS;
    WrT[j] = (c < MW + 6) ? Wr[((size_t)hi * (MW + 6) + c) * CDIM + k] : 0.0f;
  } else if (i < NCAT + NR + NWW) {
    size_t j = i - NCAT - NR;
    int hi = (int)(j / (CDIM * WCOLS));
    int r  = (int)(j % (CDIM * WCOLS));
    int k = r / WCOLS, c = r % WCOLS;
    WwT[j] = (c < 3 * MW + 6) ? Ww[((size_t)hi * (3 * MW + 6) + c) * CDIM + k] : 0.0f;
  }
}

__global__ void __launch_bounds__(NTHREADS)
ntm_main_kernel(const float* __restrict__ embs, const int* __restrict__ lens,
                const float* __restrict__ mem_bias,
                const float* __restrict__ b_ih, const float* __restrict__ b_hh,
                const float* __restrict__ br, const float* __restrict__ bw,
                const float* __restrict__ h0, const float* __restrict__ c0,
                const float* __restrict__ r0,
                const float* __restrict__ WcatT, const float* __restrict__ WrT,
                const float* __restrict__ WwT, float* __restrict__ memg,
                float* __restrict__ out) {
  extern __shared__ float smem[];
  float* z     = smem;                         // [BPW][ZS]   z = [x | h]
  float* gates = z + BPW * ZS;                 // [BPW][GS]
  float* cst   = gates + BPW * GS;             // [BPW][CS]
  float* reads = cst + BPW * CS;               // [NHEADS][BPW][MW]
  float* rwb   = reads + NHEADS * BPW * MW;    // [NHEADS][BPW][NSLOT]
  float* wwb   = rwb + NHEADS * BPW * NSLOT;   // [NHEADS][BPW][NSLOT]
  float* ho    = wwb + NHEADS * BPW * NSLOT;   // [BPW][HOS]
  float* simb  = ho + BPW * HOS;               // [BPW][SIMS]
  float* wgb   = simb + BPW * SIMS;            // [BPW][SIMS]

  const int tid  = threadIdx.x;
  const int wave = tid >> 5;
  const int lane = tid & 31;
  const int b0   = blockIdx.x * BPW;

  // ---- init recurrent state ----
  for (int i = tid; i < BPW * CDIM; i += NTHREADS) {
    int b = i / CDIM, j = i % CDIM;
    z[b * ZS + XDIM + j] = h0[j];
    cst[b * CS + j] = c0[j];
  }
  for (int i = tid; i < NHEADS * BPW * MW; i += NTHREADS) {
    int hi = i / (BPW * MW);
    int m = i % MW;
    reads[i] = r0[hi * MW + m];
  }
  for (int i = tid; i < NHEADS * BPW * NSLOT; i += NTHREADS) {
    rwb[i] = 0.0f;
    wwb[i] = 0.0f;
  }
  for (int i = tid; i < BPW * NSLOT * MW; i += NTHREADS) {
    int b = i / (NSLOT * MW);
    int r = i % (NSLOT * MW);
    memg[(size_t)(b0 + b) * NSLOT * MW + r] = mem_bias[r];
  }
  __syncthreads();

  const size_t OS_H = (size_t)T_STEPS * BSZ * ODIM;
  const size_t OS_C = OS_H + (size_t)BSZ * CDIM;

  for (int t = 0; t < T_STEPS; ++t) {
    // A: assemble x part of z = [emb | reads0 | reads1].
    // emb slice goes in via async global->LDS DMA (ASYNCcnt), 16B per lane.
    for (int i = tid; i < BPW * (IDIM / 4); i += NTHREADS) {
      int b = i / (IDIM / 4), jc = i % (IDIM / 4);
      const float* g = embs + ((size_t)t * BSZ + (b0 + b)) * IDIM + jc * 4;
      async_g2l_b128(g, z + b * ZS + jc * 4);
    }
    for (int i = tid; i < BPW * NHEADS * MW; i += NTHREADS) {
      int b = i / (NHEADS * MW), j = i % (NHEADS * MW);
      int hh = j / MW, m = j % MW;
      z[b * ZS + IDIM + j] = reads[(hh * BPW + b) * MW + m];
    }
    wait_async0();
    __syncthreads();

    // B: gate GEMM  gates[16,2048] = z[16,896] @ WcatT  (WMMA f32 16x16x4)
    for (int tix = wave; tix < GDIM / 16; tix += NWAVES) {
      int colbase = tix * 16;
      v8f acc = wmma_gemm(z, ZS, WcatT, GDIM, ZDIM, colbase);
      int m = lane & 15, half = lane >> 4;
      int col = colbase + m;
      float bias = b_ih[col] + b_hh[col];
      for (int v = 0; v < 8; ++v)
        gates[(v + 8 * half) * GS + col] = acc[v] + bias;
    }
    __syncthreads();

    // C: LSTM nonlinearity, write h,c and the h part of os_
    for (int i = tid; i < BPW * CDIM; i += NTHREADS) {
      int b = i / CDIM, j = i % CDIM;
      float gi = gates[b * GS + j];
      float gf = gates[b * GS + CDIM + j];
      float gg = gates[b * GS + 2 * CDIM + j];
      float go = gates[b * GS + 3 * CDIM + j];
      float cn = sigmoidf_(gf) * cst[b * CS + j] + sigmoidf_(gi) * tanhf(gg);
      float hn = sigmoidf_(go) * tanhf(cn);
      cst[b * CS + j] = cn;
      z[b * ZS + XDIM + j] = hn;
      int gb = b0 + b;
      out[((size_t)t * BSZ + gb) * ODIM + j] = hn;
      if (t == lens[gb] - 1) {
        out[OS_H + (size_t)gb * CDIM + j] = hn;
        out[OS_C + (size_t)gb * CDIM + j] = cn;
      }
    }
    __syncthreads();

    for (int hi = 0; hi < NHEADS; ++hi) {
      // D1: read-head GEMM  ho[16,80] = h @ WrT[hi]
      if (wave < RCOLS / 16) {
        int colbase = wave * 16;
        v8f acc = wmma_gemm(z + XDIM, ZS, WrT + (size_t)hi * CDIM * RCOLS,
                            RCOLS, CDIM, colbase);
        int m = lane & 15, half = lane >> 4;
        int col = colbase + m;
        float bias = (col < MW + 6) ? br[hi * (MW + 6) + col] : 0.0f;
        for (int v = 0; v < 8; ++v)
          ho[(v + 8 * half) * HOS + col] = acc[v] + bias;
      }
      __syncthreads();

      // D2: read addressing + read vector (wave-per-batch)
      {
        int b = wave;
        float* memb = memg + (size_t)(b0 + b) * NSLOT * MW;
        float* wrow = rwb + (hi * BPW + b) * NSLOT;
        address_head(memb, ho + b * HOS, wrow, simb + b * SIMS, wgb + b * SIMS);
        for (int m = lane; m < MW; m += 32) {
          float acc = 0.0f;
          for (int n = 0; n < NSLOT; ++n) acc += wrow[n] * memb[n * MW + m];
          reads[(hi * BPW + b) * MW + m] = acc;
          out[((size_t)t * BSZ + (b0 + b)) * ODIM + CDIM + hi * MW + m] = acc;
        }
      }
      __syncthreads();

      // D3: write-head GEMM  ho[16,208] = h @ WwT[hi]
      if (wave < WCOLS / 16) {
        int colbase = wave * 16;
        v8f acc = wmma_gemm(z + XDIM, ZS, WwT + (size_t)hi * CDIM * WCOLS,
                            WCOLS, CDIM, colbase);
        int m = lane & 15, half = lane >> 4;
        int col = colbase + m;
        float bias = (col < 3 * MW + 6) ? bw[hi * (3 * MW + 6) + col] : 0.0f;
        for (int v = 0; v < 8; ++v)
          ho[(v + 8 * half) * HOS + col] = acc[v] + bias;
      }
      __syncthreads();

      // D4: write addressing + erase/add memory update (wave-per-batch)
      {
        int b = wave;
        float* memb = memg + (size_t)(b0 + b) * NSLOT * MW;
        const float* hob = ho + b * HOS;
        float* w2row = wwb + (hi * BPW + b) * NSLOT;
        address_head(memb, hob, w2row, simb + b * SIMS, wgb + b * SIMS);
        float ev0 = sigmoidf_(hob[MW + 6 + lane]);
        float ev1 = sigmoidf_(hob[MW + 6 + lane + 32]);
        float av0 = hob[2 * MW + 6 + lane];
        float av1 = hob[2 * MW + 6 + lane + 32];
        for (int n = 0; n < NSLOT; ++n) {
          float w2 = w2row[n];
          float* mp = memb + n * MW;
          float m0 = mp[lane], m1 = mp[lane + 32];
          mp[lane]      = m0 * (1.0f - w2 * ev0) + w2 * av0;
          mp[lane + 32] = m1 * (1.0f - w2 * ev1) + w2 * av1;
        }
      }
      __syncthreads();
    }
  }
}

extern "C" void kernel_launch(void* const* d_in, const int* in_sizes, int n_in,
                              void* d_out, int out_size, void* d_ws, size_t ws_size,
                              hipStream_t stream) {
  const float* embs     = (const float*)d_in[0];
  const int*   lens     = (const int*)d_in[1];
  const float* mem_bias = (const float*)d_in[2];
  const float* W_ih     = (const float*)d_in[3];
  const float* W_hh     = (const float*)d_in[4];
  const float* b_ih     = (const float*)d_in[5];
  const float* b_hh     = (const float*)d_in[6];
  const float* Wr       = (const float*)d_in[7];
  const float* br       = (const float*)d_in[8];
  const float* Ww       = (const float*)d_in[9];
  const float* bw       = (const float*)d_in[10];
  const float* h0       = (const float*)d_in[11];
  const float* c0       = (const float*)d_in[12];
  const float* r0       = (const float*)d_in[13];
  float* out = (float*)d_out;
  float* ws  = (float*)d_ws;

  float* memg  = ws;                                   // [BSZ][NSLOT][MW]
  float* WcatT = memg + (size_t)BSZ * NSLOT * MW;      // [ZDIM][GDIM]
  float* WrT   = WcatT + (size_t)ZDIM * GDIM;          // [NHEADS][CDIM][RCOLS]
  float* WwT   = WrT + (size_t)NHEADS * CDIM * RCOLS;  // [NHEADS][CDIM][WCOLS]

  size_t prep_total = (size_t)ZDIM * GDIM +
                      (size_t)NHEADS * CDIM * RCOLS +
                      (size_t)NHEADS * CDIM * WCOLS;
  ntm_prep_kernel<<<(unsigned)((prep_total + 255) / 256), 256, 0, stream>>>(
      W_ih, W_hh, Wr, Ww, WcatT, WrT, WwT);

  size_t smem_floats = (size_t)BPW * ZS + BPW * GS + BPW * CS +
                       (size_t)NHEADS * BPW * MW + 2 * (size_t)NHEADS * BPW * NSLOT +
                       (size_t)BPW * HOS + 2 * (size_t)BPW * SIMS;
  ntm_main_kernel<<<NWG, NTHREADS, smem_floats * sizeof(float), stream>>>(
      embs, lens, mem_bias, b_ih, b_hh, br, bw, h0, c0, r0,
      WcatT, WrT, WwT, memg, out);
}